// SparseConv3DLayer_75213467287817
// MI455X (gfx1250) — compile-verified
//
#include <hip/hip_runtime.h>

// Sparse 3D conv (B,H,W,D,C)x(3,3,3,C,F)->(B,H,W,D,F) for
// B=2,H=96,W=128,D=32,C=16,F=16, fp32, via V_WMMA_F32_16X16X4_F32.
//
// One wave32 per pixel. Per spatial tap the wave stages the neighbor's
// 32x16 f32 depth slab into the CENTER of a 96-row LDS region whose top and
// bottom 32 rows are pre-filled with default_value. Depth realignment then
// becomes a pure (clamped, wave-uniform) address shift: out-of-range depth
// samples land in the guard rows and read default_value with NO per-lane
// predication -- A fragments are unconditional ds_load_b64.

typedef __attribute__((ext_vector_type(2))) float v2f;
typedef __attribute__((ext_vector_type(8))) float v8f;

#define Bn 2
#define Hn 96
#define Wn 128
#define Dn 32
#define Cn 16
#define Fn 16
#define NPIX (Bn * Hn * Wn)          // 24576, divisible by 4
#define WAVES_PER_BLOCK 4
#define KERN_ELEMS (27 * Cn * Fn)    // 6912 floats
#define GUARD 32                     // guard rows on each side of the slab
#define SLAB_ROWS (Dn + 2 * GUARD)   // 96 rows x 16 ch = 1536 floats = 6 KB

__device__ __forceinline__ int iclamp(int x, int lo, int hi) {
    return x < lo ? lo : (x > hi ? hi : x);
}

__global__ __launch_bounds__(128) void sparse_conv3d_wmma_kernel(
    const float* __restrict__ images,      // [B,H,W,D,C]
    const int*   __restrict__ base_plane,  // [B,H,W,1]
    const float* __restrict__ kern,        // [3,3,3,C,F]
    const float* __restrict__ defv,        // [1]
    float*       __restrict__ out)         // [B,H,W,D,F]
{
    __shared__ float lds_kern[KERN_ELEMS];                       // 27 KB
    __shared__ float lds_slab[WAVES_PER_BLOCK][SLAB_ROWS * Cn];  // 4 x 6 KB

    const int tid = threadIdx.x;
    // Cooperative preload of all 27 CxF weight tiles (6912/128 = 54 each).
    for (int t = tid; t < KERN_ELEMS; t += 128)
        lds_kern[t] = kern[t];

    const int wave = tid >> 5;
    const int lane = tid & 31;
    const int p    = blockIdx.x * WAVES_PER_BLOCK + wave;   // pixel id
    const int b    = p / (Hn * Wn);
    const int hw   = p - b * (Hn * Wn);
    const int h    = hw / Wn;
    const int w    = hw - h * Wn;

    const float  dv  = defv[0];
    const float4 dv4 = { dv, dv, dv, dv };
    const int   bp_c = base_plane[p];

    const int laneM = lane & 15;   // matrix row (A) / column (B,C,D)
    const int laneH = lane >> 4;   // half-wave selector

    float*  slab  = lds_slab[wave];
    float4* slab4 = (float4*)slab;

    // One-time guard fill: rows [0,32) -> float4 [0,128); rows [64,96) ->
    // float4 [256,384). Per-wave region, LDS in-order => no barrier needed.
    #pragma unroll
    for (int t = 0; t < 4; ++t) {
        slab4[lane * 4 + t]       = dv4;   // lower guard
        slab4[256 + lane * 4 + t] = dv4;   // upper guard
    }
    __syncthreads();   // for lds_kern only

    // Per-lane base address of A-fragment row d=laneM at s=0 (data starts
    // at row GUARD): floats.
    const int laneAddrF = (GUARD + laneM) * Cn + laneH * 2;

    v8f acc0 = {0.f, 0.f, 0.f, 0.f, 0.f, 0.f, 0.f, 0.f};   // d = 0..15
    v8f acc1 = {0.f, 0.f, 0.f, 0.f, 0.f, 0.f, 0.f, 0.f};   // d = 16..31

    for (int i = 0; i < 3; ++i) {
        const int  hh = h + i - 1;
        const bool vh = (hh >= 0) & (hh < Hn);
        const int  hc = iclamp(hh, 0, Hn - 1);
        for (int j = 0; j < 3; ++j) {
            const int  ww  = w + j - 1;
            const bool vw  = (ww >= 0) & (ww < Wn);
            const int  wc  = iclamp(ww, 0, Wn - 1);
            const bool vhw = vh & vw;                       // wave-uniform
            const int  np  = (b * Hn + hc) * Wn + wc;       // neighbor pixel
            const int  rel = bp_c - base_plane[np];         // depth realign

            // Stage neighbor slab (32x16 f32 = 2 KB) into the data rows
            // (float4 [128,256)). vhw is uniform across the wave -> scalar
            // branch, EXEC stays all-ones on both paths.
            if (vhw) {
                const float4* src4 =
                    (const float4*)(images + (long)np * (Dn * Cn));
                #pragma unroll
                for (int t = 0; t < 4; ++t)
                    slab4[128 + lane * 4 + t] = src4[lane * 4 + t];
            } else {
                #pragma unroll
                for (int t = 0; t < 4; ++t)
                    slab4[128 + lane * 4 + t] = dv4;
            }

            #pragma unroll
            for (int k = 0; k < 3; ++k) {
                // Clamped shift: for |s|>=32 all rows map into guard (= all
                // default_value), exactly matching the reference.
                const int s = iclamp(rel + k - 1, -GUARD, GUARD);

                // A fragments, ISA 32-bit A 16x4 layout: lanes 0-15 hold
                // K=c0,c0+1; lanes 16-31 hold K=c0+2,c0+3. Unconditional
                // LDS loads; tile1 is +16 rows = +1024 bytes.
                const v2f* rowp = (const v2f*)(slab + laneAddrF + s * Cn);
                v2f a0[4], a1[4];
                #pragma unroll
                for (int q = 0; q < 4; ++q) {
                    a0[q] = rowp[2 * q];         // byte offset q*16
                    a1[q] = rowp[128 + 2 * q];   // byte offset 1024 + q*16
                }

                // B fragments, 4x16 f32: VGPR0/1 = rows c0+2*laneH (+1),
                // column = laneM.
                const int t27 = (i * 3 + j) * 3 + k;
                const float* kbase = lds_kern + t27 * (Cn * Fn)
                                   + (2 * laneH) * Fn + laneM;
                #pragma unroll
                for (int q = 0; q < 4; ++q) {
                    v2f bfrag = { kbase[q * 4 * Fn],
                                  kbase[q * 4 * Fn + Fn] };
                    acc0 = __builtin_amdgcn_wmma_f32_16x16x4_f32(
                        false, a0[q], false, bfrag, (short)0, acc0,
                        false, false);
                    acc1 = __builtin_amdgcn_wmma_f32_16x16x4_f32(
                        false, a1[q], false, bfrag, (short)0, acc1,
                        false, false);
                }
            }
        }
    }

    // C/D layout: VGPR v -> row v + 8*laneH, column laneM.
    float* outp = out + (long)p * (Dn * Fn);
    #pragma unroll
    for (int v = 0; v < 8; ++v) {
        const int dlo = v + 8 * laneH;
        outp[dlo * Fn + laneM]        = acc0[v];
        outp[(16 + dlo) * Fn + laneM] = acc1[v];
    }
}

extern "C" void kernel_launch(void* const* d_in, const int* in_sizes, int n_in,
                              void* d_out, int out_size, void* d_ws, size_t ws_size,
                              hipStream_t stream) {
    const float* images     = (const float*)d_in[0];
    const int*   base_plane = (const int*)d_in[1];
    const float* kern       = (const float*)d_in[2];
    const float* defv       = (const float*)d_in[3];
    float*       out        = (float*)d_out;

    dim3 grid(NPIX / WAVES_PER_BLOCK);   // 6144 blocks
    dim3 block(128);                     // 4 waves of 32
    hipLaunchKernelGGL(sparse_conv3d_wmma_kernel, grid, block, 0, stream,
                       images, base_plane, kern, defv, out);
}